// PhaseAutoEncoder_61186104099313
// MI455X (gfx1250) — compile-verified
//
#include <hip/hip_runtime.h>
#include <hip/hip_bf16.h>
#include <math.h>

// ---------------------------------------------------------------------------
// PhaseAutoEncoder for MI455X (gfx1250, wave32).
// HBM-bound: 800MB read of x (k1), 800MB write of reconstructed (k5).
// k5 maps rank-2+bias reconstruction onto V_WMMA_F32_16X16X4_F32 with 100%
// output-tile utilization: A-row = [e0 e1 1 0], B = [Wd0;Wd1;bd;0].
// Specialized k5 (D known): B register-resident, fully unrolled tile loop,
// stores = 8 base pointers + immediate offsets (no per-tile address ALU).
// ---------------------------------------------------------------------------

typedef __attribute__((ext_vector_type(2))) float v2f;
typedef __attribute__((ext_vector_type(8))) float v8f;

#define NCELL 16
#define DMAX 1024
#define MAX_RANGE_F 5.6548667764616278f /* 2*pi*0.9 */
#define EXPANSION_F 2.0f

// workspace layout (float indices), specialized for N=200000, D=1000, C=16
#define OFF_COEF  0          /* C*2*D = 32000 : (1+scale_c[d])*We[d][j], planar */
#define OFF_OFFT  32000      /* C*D   = 16000 : add_c[d] + gb[d] */
#define OFF_CONST 48000      /* C*2 */
#define OFF_CNT   48032      /* C */
#define OFF_SX    48048      /* C */
#define OFF_SY    48064      /* C */
#define OFF_MAXK  48080      /* C (uint ordered keys) */
#define OFF_MINK  48096      /* C (uint ordered keys) */
#define OFF_AM    48112      /* C */
#define OFF_RATIO 48128      /* C */
#define OFF_KEEP  48144      /* C */
#define OFF_PCN   48160      /* N*2 */
#define OFF_DEV   448160     /* N   */

__device__ __forceinline__ unsigned ordKey(float f) {
    unsigned b = __float_as_uint(f);
    return (b & 0x80000000u) ? ~b : (b | 0x80000000u);
}
__device__ __forceinline__ float ordDecode(unsigned k) {
    unsigned b = (k & 0x80000000u) ? (k & 0x7FFFFFFFu) : ~k;
    return __uint_as_float(b);
}

// --- k0: per-celltype coefficient planes + offsets; init accumulators -------
__global__ void k0_precompute(const float* __restrict__ emb,
                              const float* __restrict__ Ws, const float* __restrict__ bs,
                              const float* __restrict__ Wa, const float* __restrict__ ba,
                              const float* __restrict__ gb, const float* __restrict__ We,
                              float* __restrict__ ws, int d) {
    int idx = blockIdx.x * blockDim.x + threadIdx.x;
    if (idx < NCELL) {  // re-init every launch (graph replay)
        ws[OFF_CNT + idx] = 0.0f;
        ws[OFF_SX + idx]  = 0.0f;
        ws[OFF_SY + idx]  = 0.0f;
        ((unsigned*)ws)[OFF_MAXK + idx] = 0u;
        ((unsigned*)ws)[OFF_MINK + idx] = 0xFFFFFFFFu;
    }
    int total = NCELL * d;
    if (idx >= total) return;
    int c = idx / d, dd = idx - c * d;
    float e0 = emb[c * 4 + 0], e1 = emb[c * 4 + 1];
    float e2 = emb[c * 4 + 2], e3 = emb[c * 4 + 3];
    float scale = e0 * Ws[0 * d + dd] + e1 * Ws[1 * d + dd] +
                  e2 * Ws[2 * d + dd] + e3 * Ws[3 * d + dd] + bs[dd];
    float addv  = e0 * Wa[0 * d + dd] + e1 * Wa[1 * d + dd] +
                  e2 * Wa[2 * d + dd] + e3 * Wa[3 * d + dd] + ba[dd];
    float m = 1.0f + scale;
    ws[OFF_COEF + (size_t)(2 * c + 0) * d + dd] = m * We[dd * 2 + 0];
    ws[OFF_COEF + (size_t)(2 * c + 1) * d + dd] = m * We[dd * 2 + 1];
    ws[OFF_OFFT + (size_t)c * d + dd] = addv + gb[dd];
}

// --- k0b: const_c[j] = sum_d off[c][d]*We[d][j] + be[j] ---------------------
__global__ void k0b_const(const float* __restrict__ We, const float* __restrict__ be,
                          float* __restrict__ ws, int d) {
    __shared__ float r0[256], r1[256];
    int c = blockIdx.x, tid = threadIdx.x;
    const float* off = ws + OFF_OFFT + (size_t)c * d;
    float s0 = 0.0f, s1 = 0.0f;
    for (int dd = tid; dd < d; dd += blockDim.x) {
        float o = off[dd];
        s0 += o * We[dd * 2 + 0];
        s1 += o * We[dd * 2 + 1];
    }
    r0[tid] = s0; r1[tid] = s1;
    __syncthreads();
    for (int s = 128; s > 0; s >>= 1) {
        if (tid < s) { r0[tid] += r0[tid + s]; r1[tid] += r1[tid + s]; }
        __syncthreads();
    }
    if (tid == 0) {
        ws[OFF_CONST + 2 * c + 0] = r0[0] + be[0];
        ws[OFF_CONST + 2 * c + 1] = r1[0] + be[1];
    }
}

// --- k1: big read pass; one wave per row dot-product; segment sums ----------
__global__ void k1_pc(const float* __restrict__ x, const int* __restrict__ ct,
                      float* __restrict__ ws, int n, int d) {
    __shared__ float scnt[NCELL], ssx[NCELL], ssy[NCELL];
    if (threadIdx.x < NCELL) {
        scnt[threadIdx.x] = 0.0f; ssx[threadIdx.x] = 0.0f; ssy[threadIdx.x] = 0.0f;
    }
    __syncthreads();
    const float* coef = ws + OFF_COEF;
    const float* cst  = ws + OFF_CONST;
    float* pcn = ws + OFF_PCN;
    int lane = threadIdx.x & 31;
    int wave = (blockIdx.x * blockDim.x + threadIdx.x) >> 5;
    int nwaves = (gridDim.x * blockDim.x) >> 5;
    int nvec = d >> 2;  // 250 float4s per row
    for (int i = wave; i < n; i += nwaves) {
        int c = ct[i];
        const float4* xr = (const float4*)(x + (size_t)i * d);
        const float4* c0 = (const float4*)(coef + (size_t)(2 * c + 0) * d);
        const float4* c1 = (const float4*)(coef + (size_t)(2 * c + 1) * d);
        float s0 = 0.0f, s1 = 0.0f;
        for (int k = lane; k < nvec; k += 32) {
            __builtin_prefetch(xr + k + 64, 0, 1);  // global_prefetch_b8
            float4 xv = xr[k];
            float4 a  = c0[k];
            float4 b  = c1[k];
            s0 += xv.x * a.x + xv.y * a.y + xv.z * a.z + xv.w * a.w;
            s1 += xv.x * b.x + xv.y * b.y + xv.z * b.z + xv.w * b.w;
        }
        for (int off = 16; off > 0; off >>= 1) {
            s0 += __shfl_xor(s0, off, 32);
            s1 += __shfl_xor(s1, off, 32);
        }
        if (lane == 0) {
            float p0 = s0 + cst[2 * c + 0];
            float p1 = s1 + cst[2 * c + 1];
            float nr = sqrtf(p0 * p0 + p1 * p1) + 1e-8f;
            float px = p0 / nr, py = p1 / nr;
            pcn[2 * (size_t)i + 0] = px;
            pcn[2 * (size_t)i + 1] = py;
            atomicAdd(&scnt[c], 1.0f);
            atomicAdd(&ssx[c], px);
            atomicAdd(&ssy[c], py);
        }
    }
    __syncthreads();
    if (threadIdx.x < NCELL) {
        atomicAdd(ws + OFF_CNT + threadIdx.x, scnt[threadIdx.x]);
        atomicAdd(ws + OFF_SX  + threadIdx.x, ssx[threadIdx.x]);
        atomicAdd(ws + OFF_SY  + threadIdx.x, ssy[threadIdx.x]);
    }
}

// --- k2: circular mean angle per celltype -----------------------------------
__global__ void k2_anglemean(float* __restrict__ ws) {
    int c = threadIdx.x;
    if (c < NCELL) ws[OFF_AM + c] = atan2f(ws[OFF_SY + c], ws[OFF_SX + c]);
}

// --- k3: per-point deviation (wrapped) + segment max/min --------------------
__global__ void k3_dev(const int* __restrict__ ct, float* __restrict__ ws, int n) {
    __shared__ unsigned smax[NCELL], smin[NCELL];
    if (threadIdx.x < NCELL) { smax[threadIdx.x] = 0u; smin[threadIdx.x] = 0xFFFFFFFFu; }
    __syncthreads();
    int i = blockIdx.x * blockDim.x + threadIdx.x;
    if (i < n) {
        int c = ct[i];
        float px = ws[OFF_PCN + 2 * (size_t)i + 0];
        float py = ws[OFF_PCN + 2 * (size_t)i + 1];
        float dv = atan2f(py, px) - ws[OFF_AM + c];
        dv = atan2f(sinf(dv), cosf(dv));  // wrap to (-pi, pi]
        ws[OFF_DEV + i] = dv;
        unsigned k = ordKey(dv);
        atomicMax(&smax[c], k);
        atomicMin(&smin[c], k);
    }
    __syncthreads();
    if (threadIdx.x < NCELL) {
        atomicMax((unsigned*)ws + OFF_MAXK + threadIdx.x, smax[threadIdx.x]);
        atomicMin((unsigned*)ws + OFF_MINK + threadIdx.x, smin[threadIdx.x]);
    }
}

// --- k4: expansion ratio + keep mask per celltype ---------------------------
__global__ void k4_ratio(float* __restrict__ ws) {
    int c = threadIdx.x;
    if (c < NCELL) {
        float mx = ordDecode(((unsigned*)ws)[OFF_MAXK + c]);
        float mn = ordDecode(((unsigned*)ws)[OFF_MINK + c]);
        float rng = mx - mn;
        float ratio = 1.0f;
        if (rng > 0.0f) ratio = fminf(MAX_RANGE_F, rng * EXPANSION_F) / rng;
        ws[OFF_RATIO + c] = ratio;
        ws[OFF_KEEP + c]  = (ws[OFF_CNT + c] <= 2.0f) ? 1.0f : 0.0f;
    }
}

// --- shared helper: compute expanded coords for one row ---------------------
__device__ __forceinline__ void expand_row(const float* __restrict__ ws, int c, int r,
                                           float& ex, float& ey) {
    float px = ws[OFF_PCN + 2 * (size_t)r + 0];
    float py = ws[OFF_PCN + 2 * (size_t)r + 1];
    float ang = ws[OFF_AM + c] + ws[OFF_DEV + r] * ws[OFF_RATIO + c];
    ex = cosf(ang); ey = sinf(ang);
    if (ws[OFF_KEEP + c] != 0.0f) { ex = px; ey = py; }
}

// --- k5 (specialized D): WMMA reconstruction, B register-resident -----------
// One wave per 16-row group. A (16x4 f32): lanes 0-15 = (e0,e1) of row M=lane
// (K0,K1); lanes 16-31 = (1,0) (K2 multiplies bd-row of B, K3 zero).
// B tiles preloaded into VGPRs once per wave; tile loop fully unrolled so all
// store addresses are 8 per-group base pointers + immediate offsets.
template <int DC>
__global__ void k5_wmma_spec(const float* __restrict__ ws, const int* __restrict__ ct,
                             const float* __restrict__ Wd, const float* __restrict__ bd,
                             float* __restrict__ out_exp, float* __restrict__ out_rec,
                             int gfull) {
    constexpr int NT   = DC / 16;                 // full 16-col tiles
    constexpr int TAIL = DC % 16;
    constexpr int NTT  = NT + (TAIL ? 1 : 0);

    int lane = threadIdx.x & 31;
    int l15  = lane & 15;
    int hi   = lane >> 4;                         // 0: lanes 0-15, 1: lanes 16-31
    const float* pb0 = hi ? bd : Wd;              // B row K=0 (lo) / K=2 (hi)
    float b1m = hi ? 0.0f : 1.0f;                 // mask for B row K=1 / K=3

    // --- preload all B tiles into registers (once per wave) ---
    float b0v[NTT], b1v[NTT];
#pragma unroll
    for (int t = 0; t < NT; ++t) {
        int cl = t * 16 + l15;
        b0v[t] = pb0[cl];
        b1v[t] = b1m * Wd[DC + cl];
    }
    if (TAIL) {
        int cl = NT * 16 + l15;
        bool v = cl < DC;
        b0v[NT] = v ? pb0[cl] : 0.0f;
        b1v[NT] = v ? (b1m * Wd[DC + cl]) : 0.0f;
    }

    int wave = (blockIdx.x * blockDim.x + threadIdx.x) >> 5;
    int nwaves = (gridDim.x * blockDim.x) >> 5;

    for (int g = wave; g < gfull; g += nwaves) {
        int row0 = g << 4;
        int r = row0 + l15;                       // valid row for all lanes
        int c = ct[r];
        float ex, ey;
        expand_row(ws, c, r, ex, ey);
        if (hi == 0) {
            float2 e2v; e2v.x = ex; e2v.y = ey;
            ((float2*)out_exp)[r] = e2v;
        }
        v2f A;
        A[0] = hi ? 1.0f : ex;
        A[1] = hi ? 0.0f : ey;

        // 8 per-group base pointers (one per D-matrix VGPR row)
        float* p0 = out_rec + (size_t)(row0 + hi * 8) * DC + l15;
        float* p1 = p0 + 1 * DC;
        float* p2 = p0 + 2 * DC;
        float* p3 = p0 + 3 * DC;
        float* p4 = p0 + 4 * DC;
        float* p5 = p0 + 5 * DC;
        float* p6 = p0 + 6 * DC;
        float* p7 = p0 + 7 * DC;

#pragma unroll
        for (int t = 0; t < NT; ++t) {
            v2f B; B[0] = b0v[t]; B[1] = b1v[t];
            v8f acc = {};
            v8f D = __builtin_amdgcn_wmma_f32_16x16x4_f32(
                false, A, false, B, (short)0, acc, false, false);
            p0[t * 16] = D[0]; p1[t * 16] = D[1];
            p2[t * 16] = D[2]; p3[t * 16] = D[3];
            p4[t * 16] = D[4]; p5[t * 16] = D[5];
            p6[t * 16] = D[6]; p7[t * 16] = D[7];
        }
        if (TAIL) {
            v2f B; B[0] = b0v[NT]; B[1] = b1v[NT];
            v8f acc = {};
            v8f D = __builtin_amdgcn_wmma_f32_16x16x4_f32(
                false, A, false, B, (short)0, acc, false, false);
            if (NT * 16 + l15 < DC) {             // guard stores only
                p0[NT * 16] = D[0]; p1[NT * 16] = D[1];
                p2[NT * 16] = D[2]; p3[NT * 16] = D[3];
                p4[NT * 16] = D[4]; p5[NT * 16] = D[5];
                p6[NT * 16] = D[6]; p7[NT * 16] = D[7];
            }
        }
    }
}

// --- k5 generic-D fallback: LDS-sourced B, dynamic tile loop ----------------
__global__ void k5_wmma_gen(const float* __restrict__ ws, const int* __restrict__ ct,
                            const float* __restrict__ Wd, const float* __restrict__ bd,
                            float* __restrict__ out_exp, float* __restrict__ out_rec,
                            int d, int gfull) {
    __shared__ float sWd0[DMAX], sWd1[DMAX], sBd[DMAX];
    for (int i = threadIdx.x; i < DMAX; i += blockDim.x) {
        sWd0[i] = (i < d) ? Wd[i]     : 0.0f;
        sWd1[i] = (i < d) ? Wd[d + i] : 0.0f;
        sBd[i]  = (i < d) ? bd[i]     : 0.0f;
    }
    __syncthreads();

    int lane = threadIdx.x & 31;
    int l15  = lane & 15;
    int hi   = lane >> 4;
    const float* pb0 = hi ? sBd : sWd0;
    float b1m = hi ? 0.0f : 1.0f;

    int wave = (blockIdx.x * blockDim.x + threadIdx.x) >> 5;
    int nwaves = (gridDim.x * blockDim.x) >> 5;
    int dfull = d >> 4;
    int dtail = d & 15;

    for (int g = wave; g < gfull; g += nwaves) {
        int row0 = g << 4;
        int r = row0 + l15;
        int c = ct[r];
        float ex, ey;
        expand_row(ws, c, r, ex, ey);
        if (hi == 0) {
            float2 e2v; e2v.x = ex; e2v.y = ey;
            ((float2*)out_exp)[r] = e2v;
        }
        v2f A;
        A[0] = hi ? 1.0f : ex;
        A[1] = hi ? 0.0f : ey;

        int rowbase = (row0 + hi * 8) * d + l15;
        for (int t = 0; t < dfull; ++t) {
            int cl = (t << 4) + l15;
            v2f B;
            B[0] = pb0[cl];
            B[1] = b1m * sWd1[cl];
            v8f acc = {};
            v8f D = __builtin_amdgcn_wmma_f32_16x16x4_f32(
                false, A, false, B, (short)0, acc, false, false);
            int off = rowbase + (t << 4);
#pragma unroll
            for (int r8 = 0; r8 < 8; ++r8) out_rec[off + r8 * d] = D[r8];
        }
        if (dtail) {
            int col0 = dfull << 4;
            int cl = col0 + l15;
            v2f B;
            B[0] = pb0[cl];
            B[1] = b1m * sWd1[cl];
            v8f acc = {};
            v8f D = __builtin_amdgcn_wmma_f32_16x16x4_f32(
                false, A, false, B, (short)0, acc, false, false);
            if (cl < d) {
                int off = rowbase + col0;
#pragma unroll
                for (int r8 = 0; r8 < 8; ++r8) out_rec[off + r8 * d] = D[r8];
            }
        }
    }
}

// --- k5t: scalar fallback for the (rare) row tail, rows [rstart, n) ---------
__global__ void k5_tail(const float* __restrict__ ws, const int* __restrict__ ct,
                        const float* __restrict__ Wd, const float* __restrict__ bd,
                        float* __restrict__ out_exp, float* __restrict__ out_rec,
                        int n, int d, int rstart) {
    int idx = blockIdx.x * blockDim.x + threadIdx.x;
    int total = (n - rstart) * d;
    if (idx >= total) return;
    int rr = idx / d, dd = idx - rr * d;
    int r = rstart + rr;
    int c = ct[r];
    float ex, ey;
    expand_row(ws, c, r, ex, ey);
    if (dd == 0) { out_exp[2 * (size_t)r + 0] = ex; out_exp[2 * (size_t)r + 1] = ey; }
    out_rec[(size_t)r * d + dd] = ex * Wd[dd] + ey * Wd[d + dd] + bd[dd];
}

extern "C" void kernel_launch(void* const* d_in, const int* in_sizes, int n_in,
                              void* d_out, int out_size, void* d_ws, size_t ws_size,
                              hipStream_t stream) {
    const float* x   = (const float*)d_in[0];
    const int*   ct  = (const int*)  d_in[1];
    const float* emb = (const float*)d_in[2];
    const float* Ws  = (const float*)d_in[3];
    const float* bs  = (const float*)d_in[4];
    const float* Wa  = (const float*)d_in[5];
    const float* ba  = (const float*)d_in[6];
    const float* gb  = (const float*)d_in[7];
    const float* We  = (const float*)d_in[8];
    const float* be  = (const float*)d_in[9];
    const float* Wd  = (const float*)d_in[10];
    const float* bd  = (const float*)d_in[11];

    int n = in_sizes[1];  // N = len(celltype_indices)
    int d = in_sizes[4];  // D = len(bs)

    float* wsf = (float*)d_ws;
    float* out_exp = (float*)d_out;             // expanded  [N,2]
    float* out_rec = out_exp + (size_t)n * 2;   // reconstructed [N,D]

    int pre_blocks = (NCELL * d + 255) / 256;
    k0_precompute<<<pre_blocks, 256, 0, stream>>>(emb, Ws, bs, Wa, ba, gb, We, wsf, d);
    k0b_const<<<NCELL, 256, 0, stream>>>(We, be, wsf, d);
    k1_pc<<<1280, 256, 0, stream>>>(x, ct, wsf, n, d);
    k2_anglemean<<<1, 32, 0, stream>>>(wsf);
    k3_dev<<<(n + 255) / 256, 256, 0, stream>>>(ct, wsf, n);
    k4_ratio<<<1, 32, 0, stream>>>(wsf);

    int gfull = n >> 4;                         // full 16-row groups
    if (gfull > 0) {
        int blocks5 = (gfull * 32 + 255) / 256; // exactly one wave per group
        if (d == 1000) {
            k5_wmma_spec<1000><<<blocks5, 256, 0, stream>>>(wsf, ct, Wd, bd,
                                                            out_exp, out_rec, gfull);
        } else {
            k5_wmma_gen<<<blocks5, 256, 0, stream>>>(wsf, ct, Wd, bd,
                                                     out_exp, out_rec, d, gfull);
        }
    }
    int rstart = gfull << 4;
    if (rstart < n) {
        int tail_elems = (n - rstart) * d;
        k5_tail<<<(tail_elems + 255) / 256, 256, 0, stream>>>(wsf, ct, Wd, bd,
                                                              out_exp, out_rec, n, d, rstart);
    }
}